// Seq2SeqAttentionDecoder_38826504356471
// MI455X (gfx1250) — compile-verified
//
#include <hip/hip_runtime.h>
#include <hip/hip_bf16.h>

#define VOCAB 32000
#define EMBED 512
#define HID   1024
#define NB    64      // batch
#define TDEC  64
#define SENC  128

// gfx1250 async global->LDS path (ASYNCcnt). Fallback: register prefetch.
#define USE_ASYNC_LDS 1

typedef float v2f __attribute__((ext_vector_type(2)));
typedef float v8f __attribute__((ext_vector_type(8)));

// ---------------------------------------------------------------------------
// fp32 WMMA tiled GEMM:  C(MxN) = A(MxK) * W(NxK)^T + bias
// Tile BTxBT, 256 threads = 8 waves in 4(m) x 2(n) grid.
// Wave computes (BT/4)x(BT/2) outputs = MS x NS subtiles of 16x16.
// K-panel = 16 (4 WMMA k-steps), double-buffered LDS, 1 barrier/panel,
// next panel loaded asynchronously into LDS while current one is computed.
// A fragment (16x4 f32): lanes 0-15 -> M=0..15 K={0,1}; lanes 16-31 K={2,3}.
// B fragment (4x16):     lanes 0-15 -> N=0..15 K={0,1}; lanes 16-31 K={2,3}.
// C tile: VGPR r: lanes 0-15 M=r, lanes 16-31 M=r+8; N = lane&15.
// logits_store==1 remaps row m=t*NB+b to out[b][t][*] (B,T,V layout).
// ---------------------------------------------------------------------------
template <int BT>
__global__ void __launch_bounds__(256)
gemm_wmma(const float* __restrict__ A, const float* __restrict__ W,
          const float* __restrict__ bias, float* __restrict__ C,
          int N, int K, int logits_store)
{
  constexpr int MS   = BT / 64;   // m-subtiles per wave
  constexpr int NS   = BT / 32;   // n-subtiles per wave
  constexpr int L4   = BT / 64;   // float4 loads per thread per matrix
  constexpr int ROWF = 20;        // padded LDS row (floats) -> conflict-free

  __shared__ float As[2][BT * ROWF];
  __shared__ float Ws[2][BT * ROWF];

  const int tid  = threadIdx.x;
  const int lane = tid & 31;
  const int wave = tid >> 5;
  const int wm   = wave & 3;            // 0..3
  const int wn   = wave >> 2;           // 0..1
  const int fm   = lane & 15;
  const int klo  = (lane >> 4) << 1;    // 0 or 2
  const int mstripe = wm * (BT / 4);
  const int nstripe = wn * (BT / 2);

  const int mbase = blockIdx.y * BT;
  const int nbase = blockIdx.x * BT;

  v8f acc[MS][NS];
#pragma unroll
  for (int im = 0; im < MS; ++im)
#pragma unroll
    for (int jn = 0; jn < NS; ++jn)
      acc[im][jn] = (v8f){0.f, 0.f, 0.f, 0.f, 0.f, 0.f, 0.f, 0.f};

  // per-thread panel-load slots: thread covers L4 float4s of each matrix
  int rowL[L4], colL[L4];
#pragma unroll
  for (int i = 0; i < L4; ++i) {
    const int idx = tid + i * 256;
    rowL[i] = idx >> 2;          // 0..BT-1
    colL[i] = (idx & 3) << 2;    // 0,4,8,12
  }

  auto compute_panel = [&](int buf) {
    const float* ab = &As[buf][(mstripe + fm) * ROWF + klo];
    const float* wb = &Ws[buf][(nstripe + fm) * ROWF + klo];
#pragma unroll
    for (int kk = 0; kk < 16; kk += 4) {
      v2f a[MS], b[NS];
#pragma unroll
      for (int im = 0; im < MS; ++im) {
        a[im].x = ab[im * 16 * ROWF + kk];
        a[im].y = ab[im * 16 * ROWF + kk + 1];
      }
#pragma unroll
      for (int jn = 0; jn < NS; ++jn) {
        b[jn].x = wb[jn * 16 * ROWF + kk];
        b[jn].y = wb[jn * 16 * ROWF + kk + 1];
      }
#pragma unroll
      for (int im = 0; im < MS; ++im)
#pragma unroll
        for (int jn = 0; jn < NS; ++jn)
          acc[im][jn] = __builtin_amdgcn_wmma_f32_16x16x4_f32(
              false, a[im], false, b[jn], (short)0, acc[im][jn], false, false);
    }
  };

#if USE_ASYNC_LDS
  auto issue_panel = [&](int k0, int buf) {
#pragma unroll
    for (int i = 0; i < L4; ++i) {
      const float* ga = A + (size_t)(mbase + rowL[i]) * K + k0 + colL[i];
      const unsigned la =
          (unsigned)(unsigned long long)&As[buf][rowL[i] * ROWF + colL[i]];
      asm volatile("global_load_async_to_lds_b128 %0, %1, off"
                   :: "v"(la), "v"(ga) : "memory");
      const float* gw = W + (size_t)(nbase + rowL[i]) * K + k0 + colL[i];
      const unsigned lw =
          (unsigned)(unsigned long long)&Ws[buf][rowL[i] * ROWF + colL[i]];
      asm volatile("global_load_async_to_lds_b128 %0, %1, off"
                   :: "v"(lw), "v"(gw) : "memory");
    }
  };

  issue_panel(0, 0);
  for (int k0 = 0; k0 < K; k0 += 16) {
    const int buf = (k0 >> 4) & 1;
    // current panel's async writes complete (own), then workgroup-wide:
    asm volatile("s_wait_asynccnt 0" ::: "memory");
    __syncthreads();
    // prefetch next panel into the other buffer; prior reads of it drained
    // at the barrier above (each wave's dscnt==0 before signaling).
    if (k0 + 16 < K) issue_panel(k0 + 16, buf ^ 1);
    compute_panel(buf);
  }
#else
  float4 ra[L4], rw[L4];
  auto load_regs = [&](int k0) {
#pragma unroll
    for (int i = 0; i < L4; ++i) {
      ra[i] = *(const float4*)(A + (size_t)(mbase + rowL[i]) * K + k0 + colL[i]);
      rw[i] = *(const float4*)(W + (size_t)(nbase + rowL[i]) * K + k0 + colL[i]);
    }
  };
  auto store_regs = [&](int buf) {
#pragma unroll
    for (int i = 0; i < L4; ++i) {
      *(float4*)&As[buf][rowL[i] * ROWF + colL[i]] = ra[i];
      *(float4*)&Ws[buf][rowL[i] * ROWF + colL[i]] = rw[i];
    }
  };

  load_regs(0);
  for (int k0 = 0; k0 < K; k0 += 16) {
    const int buf = (k0 >> 4) & 1;
    store_regs(buf);
    if (k0 + 16 < K) load_regs(k0 + 16);
    __syncthreads();
    compute_panel(buf);
  }
#endif

#pragma unroll
  for (int im = 0; im < MS; ++im) {
    const int m0 = mbase + mstripe + im * 16 + (lane >> 4) * 8;
#pragma unroll
    for (int jn = 0; jn < NS; ++jn) {
      const int n = nbase + nstripe + jn * 16 + fm;
      const float bv = bias ? bias[n] : 0.f;
#pragma unroll
      for (int r = 0; r < 8; ++r) {
        const int m = m0 + r;
        const float c = acc[im][jn][r] + bv;
        if (!logits_store) {
          C[(size_t)m * N + n] = c;
        } else {
          const int bb = m & (NB - 1);
          const int tt = m >> 6;
          C[(size_t)bb * TDEC * VOCAB + (size_t)tt * VOCAB + n] = c;
        }
      }
    }
  }
}

// ---------------------------------------------------------------------------
// X_emb[t][b][e] = embedding[X[b][t]][e]   (T,B,E layout)
// ---------------------------------------------------------------------------
__global__ void embed_kernel(const int* __restrict__ X,
                             const float* __restrict__ emb,
                             float* __restrict__ xemb)
{
  const int idx = blockIdx.x * blockDim.x + threadIdx.x;  // T*B*E total
  const int e  = idx % EMBED;
  const int tb = idx / EMBED;
  const int b  = tb % NB;
  const int t  = tb / NB;
  const int tok = X[b * TDEC + t];
  xemb[idx] = emb[(size_t)tok * EMBED + e];
}

// ---------------------------------------------------------------------------
// Additive attention for one step; one block per batch row b.
// scores[s] = sum_h tanh(q[h]+keys[s][h]) * wv[h]; masked softmax;
// context[h] = sum_s attn[s]*enc[s][h]; writes inp = [context, x_t].
// ---------------------------------------------------------------------------
__global__ void __launch_bounds__(256)
attn_step_kernel(const float* __restrict__ qproj, const float* __restrict__ keysp,
                 const float* __restrict__ wv, const int* __restrict__ vlen,
                 const float* __restrict__ enc, const float* __restrict__ xemb,
                 float* __restrict__ inp, int t)
{
  const int b    = blockIdx.x;
  const int tid  = threadIdx.x;
  const int lane = tid & 31;
  const int wave = tid >> 5;

  __shared__ float sc[SENC];
  __shared__ float snorm;

  const float* q  = qproj + (size_t)b * HID;
  const float* kb = keysp + (size_t)b * SENC * HID;
  const float* eb = enc   + (size_t)b * SENC * HID;

  for (int s = wave; s < SENC; s += 8) {
    float p = 0.f;
    const float* krow = kb + (size_t)s * HID;
    for (int h = lane; h < HID; h += 32)
      p += tanhf(q[h] + krow[h]) * wv[h];
    for (int off = 16; off > 0; off >>= 1)
      p += __shfl_down(p, off, 32);
    if (lane == 0) sc[s] = p;
  }
  __syncthreads();

  if (tid == 0) {
    int L = vlen[b];
    if (L < 1) L = 1;
    if (L > SENC) L = SENC;
    float mx = -3.4e38f;
    for (int s = 0; s < L; ++s) mx = fmaxf(mx, sc[s]);
    float sum = 0.f;
    for (int s = 0; s < L; ++s) { float e = expf(sc[s] - mx); sc[s] = e; sum += e; }
    for (int s = L; s < SENC; ++s) sc[s] = 0.f;
    snorm = 1.f / sum;
  }
  __syncthreads();

  const float inv = snorm;
  float* ib = inp + (size_t)b * (HID + EMBED);
  for (int h = tid; h < HID; h += 256) {
    float acc = 0.f;
#pragma unroll 4
    for (int s = 0; s < SENC; ++s) acc += sc[s] * eb[(size_t)s * HID + h];
    ib[h] = acc * inv;
  }
  const float* xr = xemb + ((size_t)t * NB + b) * EMBED;
  for (int e = tid; e < EMBED; e += 256) ib[HID + e] = xr[e];
}

// ---------------------------------------------------------------------------
// GRU combine: r,z,n gates -> h' ; biases already folded into gi/gh GEMMs.
// ---------------------------------------------------------------------------
__global__ void gru_combine_kernel(const float* __restrict__ gi,
                                   const float* __restrict__ gh,
                                   const float* __restrict__ hprev,
                                   float* __restrict__ hnew,
                                   float* __restrict__ out_t)
{
  const int idx = blockIdx.x * blockDim.x + threadIdx.x;  // NB*HID
  const int b = idx >> 10;
  const int j = idx & (HID - 1);
  const float* gib = gi + (size_t)b * 3 * HID;
  const float* ghb = gh + (size_t)b * 3 * HID;
  const float r = 1.f / (1.f + expf(-(gib[j] + ghb[j])));
  const float z = 1.f / (1.f + expf(-(gib[HID + j] + ghb[HID + j])));
  const float n = tanhf(gib[2 * HID + j] + r * ghb[2 * HID + j]);
  const float h = (1.f - z) * n + z * hprev[idx];
  hnew[idx] = h;
  if (out_t) out_t[idx] = h;
}

// ---------------------------------------------------------------------------
extern "C" void kernel_launch(void* const* d_in, const int* in_sizes, int n_in,
                              void* d_out, int out_size, void* d_ws, size_t ws_size,
                              hipStream_t stream)
{
  (void)in_sizes; (void)n_in; (void)out_size; (void)ws_size;

  const int*   X    = (const int*)d_in[0];
  const float* enc  = (const float*)d_in[1];
  const float* h_in = (const float*)d_in[2];
  const int*   vlen = (const int*)d_in[3];
  const float* emb  = (const float*)d_in[4];
  const float* Wq   = (const float*)d_in[5];
  const float* Wk   = (const float*)d_in[6];
  const float* wv   = (const float*)d_in[7];
  const float* Wih0 = (const float*)d_in[8];
  const float* Whh0 = (const float*)d_in[9];
  const float* bih0 = (const float*)d_in[10];
  const float* bhh0 = (const float*)d_in[11];
  const float* Wih1 = (const float*)d_in[12];
  const float* Whh1 = (const float*)d_in[13];
  const float* bih1 = (const float*)d_in[14];
  const float* bhh1 = (const float*)d_in[15];
  const float* Wd   = (const float*)d_in[16];
  const float* bd   = (const float*)d_in[17];
  float* out = (float*)d_out;

  // workspace carve-out (floats); ~60.1 MB total
  float* p = (float*)d_ws;
  float* keysp = p; p += (size_t)NB * SENC * HID;        // 8388608
  float* xemb  = p; p += (size_t)TDEC * NB * EMBED;      // 2097152
  float* outs  = p; p += (size_t)TDEC * NB * HID;        // 4194304
  float* hbuf  = p; p += (size_t)2 * NB * HID;           // 131072
  float* h0 = hbuf;
  float* h1 = hbuf + (size_t)NB * HID;
  float* qproj = p; p += (size_t)NB * HID;               // 65536
  float* inp   = p; p += (size_t)NB * (HID + EMBED);     // 98304
  float* gi0 = p; p += (size_t)NB * 3 * HID;             // 196608
  float* gh0 = p; p += (size_t)NB * 3 * HID;
  float* gi1 = p; p += (size_t)NB * 3 * HID;
  float* gh1 = p; p += (size_t)NB * 3 * HID;

  // h init (do not mutate inputs)
  hipMemcpyAsync(hbuf, h_in, sizeof(float) * 2 * NB * HID,
                 hipMemcpyDeviceToDevice, stream);

  // embedding gather (T,B,E)
  embed_kernel<<<(TDEC * NB * EMBED) / 256, 256, 0, stream>>>(X, emb, xemb);

  // keys_proj = enc @ Wk^T : (B*S=8192, H) x (H, H)^T, 128x128 tiles
  gemm_wmma<128><<<dim3(HID / 128, (NB * SENC) / 128), 256, 0, stream>>>(
      enc, Wk, nullptr, keysp, HID, HID, 0);

  for (int t = 0; t < TDEC; ++t) {
    // q = h1 @ Wq^T  (M=64 -> 64x64 tiles)
    gemm_wmma<64><<<dim3(HID / 64, NB / 64), 256, 0, stream>>>(
        h1, Wq, nullptr, qproj, HID, HID, 0);
    // attention + context + concat x_t -> inp (B, H+E)
    attn_step_kernel<<<NB, 256, 0, stream>>>(qproj, keysp, wv, vlen, enc, xemb,
                                             inp, t);
    // GRU layer 0
    gemm_wmma<64><<<dim3(3 * HID / 64, NB / 64), 256, 0, stream>>>(
        inp, Wih0, bih0, gi0, 3 * HID, HID + EMBED, 0);
    gemm_wmma<64><<<dim3(3 * HID / 64, NB / 64), 256, 0, stream>>>(
        h0, Whh0, bhh0, gh0, 3 * HID, HID, 0);
    gru_combine_kernel<<<(NB * HID) / 256, 256, 0, stream>>>(gi0, gh0, h0, h0,
                                                             nullptr);
    // GRU layer 1
    gemm_wmma<64><<<dim3(3 * HID / 64, NB / 64), 256, 0, stream>>>(
        h0, Wih1, bih1, gi1, 3 * HID, HID, 0);
    gemm_wmma<64><<<dim3(3 * HID / 64, NB / 64), 256, 0, stream>>>(
        h1, Whh1, bhh1, gh1, 3 * HID, HID, 0);
    gru_combine_kernel<<<(NB * HID) / 256, 256, 0, stream>>>(
        gi1, gh1, h1, h1, outs + (size_t)t * NB * HID);
  }

  // logits = outs @ Wd^T + bd, stored transposed to (B,T,V), 128x128 tiles
  gemm_wmma<128><<<dim3(VOCAB / 128, (TDEC * NB) / 128), 256, 0, stream>>>(
      outs, Wd, bd, out, VOCAB, HID, 1);

  // final_h = [h0, h1] appended after logits
  hipMemcpyAsync(out + (size_t)NB * TDEC * VOCAB, hbuf,
                 sizeof(float) * 2 * NB * HID, hipMemcpyDeviceToDevice, stream);
}